// autoencoder_65601330479367
// MI455X (gfx1250) — compile-verified
//
#include <hip/hip_runtime.h>
#include <hip/hip_bf16.h>
#include <math.h>

// ---------------------------------------------------------------------------
// Autoencoder on MI455X (gfx1250).
// Mat-vec layers via V_WMMA_F32_16X16X4_F32: each wave owns 16 output rows,
// A operand = dense 16x4 fp32 weight tile, B operand = x broadcast along N.
// K is a template parameter so the k-loop trip count is compile-time and the
// unroll-by-8 produces clauses of global_load_b64 with staggered waits
// (deep MLP of outstanding loads) instead of load/wait/wmma lockstep.
// Split-K=16 for wave-level parallelism; partials reduced in finalize kernel.
// ---------------------------------------------------------------------------

typedef __attribute__((ext_vector_type(2))) float v2f;
typedef __attribute__((ext_vector_type(8))) float v8f;

#define SPLIT_K 16
#define BSZ 1024

// Wave layout per V_WMMA_F32_16X16X4_F32 (ISA 7.12.2, 32-bit A 16x4):
//   lane L (L<16):  M = L,      holds K = k0+0, k0+1   (one b64 load)
//   lane L (L>=16): M = L-16,   holds K = k0+2, k0+3
// B (4x16) broadcast: lane L holds x[k0 + 2*(L>>4) + {0,1}] for every N.
// C/D (16x16 f32): lane 0 -> D[M=0..7][N=0] in acc[0..7], lane 16 -> M=8..15.
template <int K>
__global__ void __launch_bounds__(256)
matvec_wmma_kernel(const float* __restrict__ W, const float* __restrict__ x,
                   float* __restrict__ partial, int M) {
  constexpr int kLen = K / SPLIT_K;       // 192 (K=3072) or 128 (K=2048)

  const int wave   = (blockIdx.x * blockDim.x + threadIdx.x) >> 5;
  const int lane   = threadIdx.x & 31;
  const int nRG    = M >> 4;              // row groups of 16
  const int rg     = wave % nRG;
  const int kc     = wave / nRG;          // 0..SPLIT_K-1
  const int kBeg   = kc * kLen;

  const int m      = lane & 15;
  const int khalf  = (lane >> 4) << 1;    // 0 or 2

  const float* __restrict__ wp = W + (size_t)(rg * 16 + m) * (size_t)K
                                   + (size_t)(kBeg + khalf);
  const float* __restrict__ xp = x + kBeg + khalf;

  v8f acc = {};
#pragma unroll 8
  for (int k0 = 0; k0 < kLen; k0 += 4) {
    v2f a = *(const v2f*)(wp + k0);
    v2f b = *(const v2f*)(xp + k0);
    acc = __builtin_amdgcn_wmma_f32_16x16x4_f32(
        /*neg_a=*/false, a, /*neg_b=*/false, b,
        /*c_mod=*/(short)0, acc, /*reuse_a=*/false, /*reuse_b=*/false);
  }

  // Column N=0 result lives in lanes 0 (M=0..7) and 16 (M=8..15).
  if ((lane & 15) == 0) {
    const int rbase = rg * 16 + ((lane >> 4) << 3);
    float* p = partial + (size_t)kc * (size_t)M + rbase;
#pragma unroll
    for (int j = 0; j < 8; ++j) p[j] = acc[j];
  }
}

// acts: 0 = identity, 1 = tanh, 2 = tanhshrink (s - tanh(s))
__global__ void __launch_bounds__(256)
finalize_kernel(const float* __restrict__ partial, const float* __restrict__ bias,
                float* __restrict__ out, float* __restrict__ out2,
                int M, int act) {
  const int i = blockIdx.x * blockDim.x + threadIdx.x;
  if (i >= M) return;
  float s = bias[i];
#pragma unroll
  for (int k = 0; k < SPLIT_K; ++k) s += partial[(size_t)k * (size_t)M + i];
  float r;
  if (act == 1)      r = tanhf(s);
  else if (act == 2) r = s - tanhf(s);
  else               r = s;
  out[i] = r;
  if (out2) out2[i] = r;
}

// Pairwise Euclidean distances: a is [BSZ][DIMS], out is [BSZ*BSZ] flat.
template <int DIMS>
__global__ void __launch_bounds__(256)
pdist_kernel(const float* __restrict__ a, float* __restrict__ out) {
  const int idx = blockIdx.x * blockDim.x + threadIdx.x;
  const int i = idx >> 10;
  const int j = idx & (BSZ - 1);
  float s = 0.f;
#pragma unroll
  for (int c = 0; c < DIMS; ++c) {
    const float d = a[i * DIMS + c] - a[j * DIMS + c];
    s += d * d;
  }
  out[idx] = sqrtf(s);
}

extern "C" void kernel_launch(void* const* d_in, const int* in_sizes, int n_in,
                              void* d_out, int out_size, void* d_ws, size_t ws_size,
                              hipStream_t stream) {
  (void)in_sizes; (void)n_in; (void)out_size; (void)ws_size;

  const float* x   = (const float*)d_in[0];
  const float* ew1 = (const float*)d_in[1];  const float* eb1 = (const float*)d_in[2];
  const float* ew2 = (const float*)d_in[3];  const float* eb2 = (const float*)d_in[4];
  const float* ew3 = (const float*)d_in[5];  const float* eb3 = (const float*)d_in[6];
  const float* ew4 = (const float*)d_in[7];  const float* eb4 = (const float*)d_in[8];
  const float* dw1 = (const float*)d_in[9];  const float* db1 = (const float*)d_in[10];
  const float* dw2 = (const float*)d_in[11]; const float* db2 = (const float*)d_in[12];
  const float* dw3 = (const float*)d_in[13]; const float* db3 = (const float*)d_in[14];
  const float* dw4 = (const float*)d_in[15]; const float* db4 = (const float*)d_in[16];

  const int A = 3072, D = 2048;

  // Workspace layout (floats)
  float* ws   = (float*)d_ws;
  float* h1   = ws;                 // 3072
  float* h2   = ws + 3072;          // 3072
  float* h3   = ws + 6144;          // 3072
  float* y    = ws + 9216;          // 2048
  float* g1   = ws + 11264;         // 3072
  float* g2   = ws + 14336;         // 3072
  float* g3   = ws + 17408;         // 3072
  float* part = ws + 20480;         // 3072 * SPLIT_K

  // Output layout: output[3072] | in_diff[1M] | lat_diff[1M] | lat_repr[2048]
  float* out          = (float*)d_out;
  float* out_output   = out;
  float* out_in_diff  = out + A;
  float* out_lat_diff = out + A + BSZ * BSZ;
  float* out_lat_repr = out + A + 2 * BSZ * BSZ;

  auto layer3072 = [&](const float* W, const float* b, const float* in,
                       float* o, float* o2, int M, int act) {
    const int waves  = (M >> 4) * SPLIT_K;       // divisible by 8
    const int blocks = waves >> 3;               // 8 waves (256 thr) per block
    matvec_wmma_kernel<3072><<<blocks, 256, 0, stream>>>(W, in, part, M);
    finalize_kernel<<<(M + 255) / 256, 256, 0, stream>>>(part, b, o, o2, M, act);
  };
  auto layer2048 = [&](const float* W, const float* b, const float* in,
                       float* o, float* o2, int M, int act) {
    const int waves  = (M >> 4) * SPLIT_K;
    const int blocks = waves >> 3;
    matvec_wmma_kernel<2048><<<blocks, 256, 0, stream>>>(W, in, part, M);
    finalize_kernel<<<(M + 255) / 256, 256, 0, stream>>>(part, b, o, o2, M, act);
  };

  // Encoder
  layer3072(ew1, eb1, x,  h1, nullptr,      A, 1);   // tanh
  layer3072(ew2, eb2, h1, h2, nullptr,      A, 1);   // tanh
  layer3072(ew3, eb3, h2, h3, nullptr,      A, 0);   // linear
  layer3072(ew4, eb4, h3, y,  out_lat_repr, D, 2);   // tanhshrink -> y (+lat_repr)

  // Pairwise distance maps (stop_gradient — values only)
  pdist_kernel<3><<<(BSZ * BSZ) / 256, 256, 0, stream>>>(x, out_in_diff);
  pdist_kernel<2><<<(BSZ * BSZ) / 256, 256, 0, stream>>>(y, out_lat_diff);

  // Decoder (first layer has K = D = 2048)
  layer2048(dw1, db1, y,  g1, nullptr,      A, 1);   // tanh
  layer3072(dw2, db2, g1, g2, nullptr,      A, 1);   // tanh
  layer3072(dw3, db3, g2, g3, nullptr,      A, 0);   // linear
  layer3072(dw4, db4, g3, out_output, nullptr, A, 1); // final tanh -> output
}